// MultiHeadSelfAttention_42253888258520
// MI455X (gfx1250) — compile-verified
//
#include <hip/hip_runtime.h>
#include <hip/hip_bf16.h>

// ---------------------------------------------------------------------------
// MI455X (gfx1250) fused single-head causal attention, tiled x16 on features.
//
// Roofline: ~3.8 GFLOP vs a 34MB HBM floor (~1.5us at 23.3 TB/s) -> memory
// bound even at V_WMMA_F32_16X16X4_F32 rates, so we keep full fp32 reference
// fidelity. Matmuls use v_wmma_f32_16x16x4_f32; x-tile staging uses the
// Tensor Data Mover (tensor_load_to_lds + s_wait_tensorcnt) where available.
// ---------------------------------------------------------------------------

typedef float v2f __attribute__((ext_vector_type(2)));
typedef float v8f __attribute__((ext_vector_type(8)));
typedef unsigned int u32x4 __attribute__((ext_vector_type(4)));
typedef int i32x4 __attribute__((ext_vector_type(4)));
typedef int i32x8 __attribute__((ext_vector_type(8)));

__device__ __forceinline__ v8f wmma_f32(v2f a, v2f b, v8f c) {
  // 8 args: (neg_a, A, neg_b, B, c_mod, C, reuse_a, reuse_b)
  return __builtin_amdgcn_wmma_f32_16x16x4_f32(
      false, a, false, b, (short)0, c, false, false);
}

// Problem constants
#define BB 2
#define TT 2048
#define CC 1024
#define HS 64
#define BT (BB * TT)   // 4096

// TDM builtin exists on both toolchains but with different arity:
//   ROCm 7.2 / clang-22 : 5 args (g0, g1, g2, g3, cpol)
//   upstream  / clang-23: 6 args (g0, g1, g2, g3, x8, cpol)
#if defined(__has_builtin)
#if __has_builtin(__builtin_amdgcn_tensor_load_to_lds)
#define HAVE_TDM 1
#endif
#endif

#if defined(HAVE_TDM)
__device__ __forceinline__ void tdm_load_2d(u32x4 g0, i32x8 g1) {
  i32x4 gz = {0, 0, 0, 0};
#if __clang_major__ >= 23
  i32x8 gz8 = {0, 0, 0, 0, 0, 0, 0, 0};
  __builtin_amdgcn_tensor_load_to_lds(g0, g1, gz, gz, gz8, 0);
#else
  __builtin_amdgcn_tensor_load_to_lds(g0, g1, gz, gz, 0);
#endif
}
#endif

// ---------------------------------------------------------------------------
// Kernel 1: q/k/v = x @ W^T + b  (three [BT,64] fp32 matrices into workspace)
// Block: 12 waves (384 thr). Wave w -> matrix (w/4), 16-col tile (w%4).
// x tile (16 rows) staged through LDS in 256-float K-chunks via the Tensor
// Data Mover; TDM pad fields reproduce the 258-float padded row stride so the
// 16-lane A-operand ds_load_b64 stays bank-conflict-free.
// ---------------------------------------------------------------------------
__global__ __launch_bounds__(384) void qkv_proj_kernel(
    const float* __restrict__ x,
    const float* __restrict__ Wq, const float* __restrict__ bq,
    const float* __restrict__ Wk, const float* __restrict__ bk,
    const float* __restrict__ Wv, const float* __restrict__ bv,
    float* __restrict__ qo, float* __restrict__ ko, float* __restrict__ vo) {
  __shared__ float xs[16 * 258];

  const int tid  = threadIdx.x;
  const int wid  = tid >> 5;
  const int lane = tid & 31;
  const int rowbase = blockIdx.x * 16;

  const int m = wid >> 2;   // 0=q 1=k 2=v
  const int n = wid & 3;    // which 16-col tile of 64

  const float* W    = (m == 0) ? Wq : (m == 1) ? Wk : Wv;
  const float* bias = (m == 0) ? bq : (m == 1) ? bk : bv;
  float*       outp = (m == 0) ? qo : (m == 1) ? ko : vo;

  const int col = lane & 15;        // N index (B/C/D layouts)
  const int h   = lane >> 4;        // lane half
  const int kkb = h * 2;            // K sub-offset within a 4-wide WMMA step

  // B operand: B[kk][nn] = W[n*16+nn][k+kk]  -> per-lane row pointer
  const float* wrowp = W + (size_t)(n * 16 + col) * CC;

  v8f acc = {};

  for (int kc = 0; kc < CC; kc += 256) {
#if defined(HAVE_TDM)
    if (wid == 0) {
      // D# group 0: count=1 | lds_addr | global_addr(57b) | type=2
      const unsigned long long ga =
          (unsigned long long)(uintptr_t)(x + (size_t)rowbase * CC + kc);
      const unsigned int lds_byte =
          (unsigned int)(uintptr_t)(void*)xs;  // LDS aperture: addr[31:0]
      u32x4 g0;
      g0[0] = 1u;                                   // count=1, user mode
      g0[1] = lds_byte;                             // lds_addr (bytes)
      g0[2] = (unsigned int)ga;                     // global_addr[31:0]
      g0[3] = (unsigned int)((ga >> 32) & 0x01FFFFFFu) | (2u << 30);  // type=2
      // D# group 1: data_size=4B(2), pad_enable, pad_interval=7 (256 DW),
      // pad_amount=1 (2 DW); tensor 1024x4096; tile 256x16; dim0 stride 1024.
      i32x8 g1;
      g1[0] = (2 << 16) | (1 << 20) | (7 << 22) | (1 << 25);
      g1[1] = (int)(1024u << 16);   // tensor_dim0[15:0] << 16
      g1[2] = (int)(4096u << 16);   // tensor_dim0[31:16]=0 | tensor_dim1 lo
      g1[3] = (int)(256u << 16);    // tensor_dim1 hi=0 | tile_dim0=256
      g1[4] = 16;                   // tile_dim1=16, tile_dim2=0
      g1[5] = 1024;                 // tensor_dim0_stride lo
      g1[6] = 0;                    // stride hi | tensor_dim1_stride lo
      g1[7] = 0;
      tdm_load_2d(g0, g1);
      __builtin_amdgcn_s_wait_tensorcnt(0);
    }
    __syncthreads();
#else
    // Fallback: cooperative float4 load of the 16x256 x-chunk into LDS
    for (int i = tid; i < 16 * 64; i += 384) {
      const int r  = i >> 6;
      const int c4 = i & 63;
      const float4 f =
          *(const float4*)(x + (size_t)(rowbase + r) * CC + kc + c4 * 4);
      float* dst = &xs[r * 258 + c4 * 4];
      dst[0] = f.x; dst[1] = f.y; dst[2] = f.z; dst[3] = f.w;
    }
    __syncthreads();
#endif

#pragma unroll 4
    for (int k = 0; k < 256; k += 4) {
      // A: lane = M row, VGPR pair = (K+kkb, K+kkb+1)
      v2f a  = *(const v2f*)(&xs[col * 258 + k + kkb]);
      v2f bm = *(const v2f*)(wrowp + kc + k + kkb);
      acc = wmma_f32(a, bm, acc);
    }
    __syncthreads();
  }

  const float bb = bias[n * 16 + col];  // bias depends only on N
#pragma unroll
  for (int r = 0; r < 8; ++r) {
    const int trow = rowbase + r + 8 * h;   // C/D layout: VGPR r, half h
    outp[(size_t)trow * HS + n * 16 + col] = acc[r] + bb;
  }
}

// ---------------------------------------------------------------------------
// Kernel 2: flash-attention over one 16-row query tile per wave.
// 4 waves/block, grid = B * T/64 = 64. Online softmax state lives per
// (VGPR row, lane-half); 16-lane butterfly shuffles reduce each row.
// P is transposed C/D->A layout through a padded (stride-18) LDS tile.
// Output is normalized and tiled x16 across the feature dim directly.
// ---------------------------------------------------------------------------
__global__ __launch_bounds__(128) void attn_kernel(
    const float* __restrict__ q, const float* __restrict__ k,
    const float* __restrict__ v, float* __restrict__ out) {
  __shared__ float plds[4][16 * 18];

  const int tid  = threadIdx.x;
  const int wid  = tid >> 5;
  const int lane = tid & 31;

  const int b      = blockIdx.x >> 5;                  // batch (grid=64)
  const int qt     = ((blockIdx.x & 31) << 2) | wid;   // query tile 0..127
  const int qbase  = qt * 16;
  const int col    = lane & 15;
  const int h      = lane >> 4;
  const int kkb    = h * 2;
  const float scale = 0.03125f;  // C^-0.5 = 1/32 (reference scales by C!)

  // Q tile in A layout: lane=M, 16 K-chunks of 4
  const float* qrow = q + (size_t)(b * TT + qbase + col) * HS;
  v2f a[16];
#pragma unroll
  for (int c = 0; c < 16; ++c) a[c] = *(const v2f*)(qrow + 4 * c + kkb);

  v8f o[4] = {{}, {}, {}, {}};
  float mrow[8], lrow[8];
#pragma unroll
  for (int r = 0; r < 8; ++r) { mrow[r] = -3.0e38f; lrow[r] = 0.0f; }

  float* pbuf = plds[wid];

  for (int j = 0; j <= qt; ++j) {
    if (j < qt) {  // prefetch next K/V tiles -> global_prefetch_b8
      __builtin_prefetch(k + (size_t)(b * TT + (j + 1) * 16 + col) * HS, 0, 1);
      __builtin_prefetch(v + (size_t)(b * TT + (j + 1) * 16 + col) * HS, 0, 1);
    }

    // S = (Q @ K^T): B[kk][nn] = k[j*16+nn][4c+kk]
    v8f s = {};
    const float* krow = k + (size_t)(b * TT + j * 16 + col) * HS;
#pragma unroll
    for (int c = 0; c < 16; ++c) {
      v2f bm = *(const v2f*)(krow + 4 * c + kkb);
      s = wmma_f32(a[c], bm, s);
    }

    // scale + causal mask + online softmax (per row r, lane-half h)
#pragma unroll
    for (int r = 0; r < 8; ++r) {
      float sv = s[r] * scale;
      if (j == qt && col > r + 8 * h) sv = -3.0e38f;  // diagonal tile mask

      float mx = sv;
      mx = fmaxf(mx, __shfl_xor(mx, 1, 32));
      mx = fmaxf(mx, __shfl_xor(mx, 2, 32));
      mx = fmaxf(mx, __shfl_xor(mx, 4, 32));
      mx = fmaxf(mx, __shfl_xor(mx, 8, 32));

      const float mnew  = fmaxf(mrow[r], mx);
      const float alpha = __expf(mrow[r] - mnew);
      const float p     = __expf(sv - mnew);

      float sum = p;
      sum += __shfl_xor(sum, 1, 32);
      sum += __shfl_xor(sum, 2, 32);
      sum += __shfl_xor(sum, 4, 32);
      sum += __shfl_xor(sum, 8, 32);

      lrow[r] = lrow[r] * alpha + sum;
      mrow[r] = mnew;

      o[0][r] *= alpha; o[1][r] *= alpha; o[2][r] *= alpha; o[3][r] *= alpha;

      pbuf[(r + 8 * h) * 18 + col] = p;  // C/D layout -> row-major LDS
    }

    // O += P @ V  : A from LDS (lane=M=col), B[kk][nn]=v[j*16+4c+kk][d*16+nn]
#pragma unroll
    for (int c = 0; c < 4; ++c) {
      v2f pa = *(const v2f*)(&pbuf[col * 18 + 4 * c + kkb]);
      const size_t vr = (size_t)(b * TT + j * 16 + 4 * c + kkb) * HS;
#pragma unroll
      for (int d = 0; d < 4; ++d) {
        v2f bv2;
        bv2.x = v[vr + d * 16 + col];
        bv2.y = v[vr + HS + d * 16 + col];
        o[d] = wmma_f32(pa, bv2, o[d]);
      }
    }
  }

  // normalize and replicate across the 16 identical heads
#pragma unroll
  for (int r = 0; r < 8; ++r) {
    const float inv  = 1.0f / lrow[r];
    const int   trow = qbase + r + 8 * h;
    float* orow = out + (size_t)(b * TT + trow) * CC;
    float vals[4];
#pragma unroll
    for (int d = 0; d < 4; ++d) vals[d] = o[d][r] * inv;
    for (int head = 0; head < 16; ++head) {
#pragma unroll
      for (int d = 0; d < 4; ++d) orow[head * HS + d * 16 + col] = vals[d];
    }
  }
}

// ---------------------------------------------------------------------------
extern "C" void kernel_launch(void* const* d_in, const int* in_sizes, int n_in,
                              void* d_out, int out_size, void* d_ws,
                              size_t ws_size, hipStream_t stream) {
  const float* x  = (const float*)d_in[0];
  const float* Wq = (const float*)d_in[1];
  const float* bq = (const float*)d_in[2];
  const float* Wk = (const float*)d_in[3];
  const float* bk = (const float*)d_in[4];
  const float* Wv = (const float*)d_in[5];
  const float* bv = (const float*)d_in[6];

  float* qws = (float*)d_ws;             // [4096, 64]
  float* kws = qws + (size_t)BT * HS;    // [4096, 64]
  float* vws = kws + (size_t)BT * HS;    // [4096, 64]  (3 MB total)

  qkv_proj_kernel<<<BT / 16, 384, 0, stream>>>(x, Wq, bq, Wk, bk, Wv, bv,
                                               qws, kws, vws);
  attn_kernel<<<BB * (TT / 64), 128, 0, stream>>>(qws, kws, vws,
                                                  (float*)d_out);
}